// QualityGatedMamba_29858612642530
// MI455X (gfx1250) — compile-verified
//
#include <hip/hip_runtime.h>
#include <hip/hip_bf16.h>

// ---------------- problem constants ----------------
#define D_MODEL 1024
#define D_STATE 16
#define D_CONV  4
#define D_INNER 2048
#define B_SZ    2
#define T_LEN   2048
#define M_ROWS  (B_SZ * T_LEN)          // 4096

typedef float v2f __attribute__((ext_vector_type(2)));
typedef float v8f __attribute__((ext_vector_type(8)));
typedef unsigned int       u32;
typedef unsigned long long u64;
typedef u32 v4u __attribute__((ext_vector_type(4)));
typedef int  v4i __attribute__((ext_vector_type(4)));
typedef int  v8i __attribute__((ext_vector_type(8)));

// GEMM tiling
#define BM 128          // block tile M
#define BN 64           // block tile N
#define KC 32           // k-chunk staged in LDS per TDM transfer
#define A_STRIDE (KC + 2)   // LDS row stride (floats), via TDM pad: 32 DW + 2 DW
#define B_STRIDE (BN + 2)   // LDS row stride (floats), via TDM pad: 64 DW + 2 DW

// =====================================================================
// Tensor Data Mover: 2D tile (global, row-major) -> LDS, with LDS row
// padding inserted by the TDM itself (pad_enable).  D# built per ISA
// cdna5 ch.8: group0 = {count, lds_addr, global_addr, type=2},
// group1 = {data_size/pad, tensor dims, tile dims, dim0 stride}.
// Tracked by TENSORcnt.  OOB reads return zero (used for N<BN tails).
// =====================================================================
__device__ __forceinline__
void tdm_load_2d_to_lds(u32 lds_byte_off, const void* gptr,
                        u32 tensor_w, u32 tensor_h,   // elems / rows remaining
                        u32 tile_w,   u32 tile_h,     // tile elems / rows
                        u32 row_stride_elems,
                        u32 pad_interval_code,        // 4 -> every 32 DW, 5 -> every 64 DW
                        u32 pad_amount_code)          // 1 -> 2 DWORDs
{
    const u64 ga = (u64)(uintptr_t)gptr;
    v4u g0;
    g0.x = 1u;                                            // count = 1 (user D#)
    g0.y = lds_byte_off;                                  // LDS byte address
    g0.z = (u32)ga;                                       // global_addr[31:0]
    g0.w = (u32)((ga >> 32) & 0x01ffffffu) | (2u << 30);  // addr[56:32] | type=2

    const u32 w0 = (2u << 16)                // data_size = 2 -> 4 bytes
                 | (1u << 20)                // pad_enable
                 | (pad_interval_code << 22)
                 | (pad_amount_code << 25);  // workgroup_mask = 0 (no cluster)
    v8i g1;
    g1[0] = (int)w0;
    g1[1] = (int)((tensor_w & 0xffffu) << 16);                                   // dim0[15:0]
    g1[2] = (int)(((tensor_w >> 16) & 0xffffu) | ((tensor_h & 0xffffu) << 16));  // dim0[31:16]|dim1[15:0]
    g1[3] = (int)(((tensor_h >> 16) & 0xffffu) | ((tile_w & 0xffffu) << 16));    // dim1[31:16]|tile_dim0
    g1[4] = (int)(tile_h & 0xffffu);                                             // tile_dim1 (tile_dim2=0)
    g1[5] = (int)row_stride_elems;                                               // dim0_stride[31:0]
    g1[6] = 0;                                                                   // stride hi / dim1_stride lo
    g1[7] = 0;

    v4i z4 = (v4i)0;
#if __clang_major__ >= 23
    v8i z8 = (v8i)0;
    __builtin_amdgcn_tensor_load_to_lds(g0, g1, z4, z4, z8, 0);
#else
    __builtin_amdgcn_tensor_load_to_lds(g0, g1, z4, z4, 0);
#endif
}

__device__ __forceinline__ u32 lds_off_u32(const void* p)
{
    // Flat shared address: addr[31:0] is the LDS offset (ISA 10.2 aperture calc)
    return (u32)(uintptr_t)p;
}

// =====================================================================
// fp32 WMMA GEMM with TDM->LDS double buffering.
// C[M,N] = A[M,K] * B[K,N], row-major.  Block: 256 thr = 8 waves,
// block tile 128x64, wave tile 32x32 (2x2 V_WMMA_F32_16X16X4_F32).
// M % 128 == 0, K % 32 == 0 (true for all calls); N tail handled by
// TDM zero-fill on OOB reads + guarded stores (EXEC all-ones at WMMA).
// =====================================================================
__global__ __launch_bounds__(256)
void wmma_gemm_f32(const float* __restrict__ A, const float* __restrict__ B,
                   float* __restrict__ C, int M, int N, int K,
                   int lda, int ldb, int ldc)
{
    __shared__ float Abuf[2][BM * A_STRIDE];   // 2 x 17408 B
    __shared__ float Bbuf[2][KC * B_STRIDE];   // 2 x  8448 B

    const int lane = threadIdx.x & 31;
    const int wave = threadIdx.x >> 5;       // 0..7
    const int wr   = wave >> 1;              // 0..3 along M
    const int wc   = wave & 1;               // 0..1 along N
    const int mblk = blockIdx.y * BM;
    const int nblk = blockIdx.x * BN;

    const int half = lane >> 4;              // 0: lanes 0-15, 1: lanes 16-31
    const int l    = lane & 15;

    const bool issuer = (wave == 0);

    auto issue_chunk = [&](int buf, int k0) {
        // A tile: BM x KC rows from A[mblk + 0.., k0..]
        tdm_load_2d_to_lds(lds_off_u32(&Abuf[buf][0]),
                           A + (size_t)mblk * lda + k0,
                           (u32)(K - k0), (u32)(M - mblk),
                           KC, BM, (u32)lda,
                           /*interval 32 DW*/4, /*amount 2 DW*/1);
        // B tile: KC x BN from B[k0.., nblk..]; OOB cols (N tail) read as 0
        tdm_load_2d_to_lds(lds_off_u32(&Bbuf[buf][0]),
                           B + (size_t)k0 * ldb + nblk,
                           (u32)(N - nblk), (u32)(K - k0),
                           BN, KC, (u32)ldb,
                           /*interval 64 DW*/5, /*amount 2 DW*/1);
    };

    v8f c00 = {}; v8f c01 = {}; v8f c10 = {}; v8f c11 = {};

    if (issuer) issue_chunk(0, 0);

    const int nchunk = K / KC;
    for (int kc = 0; kc < nchunk; ++kc) {
        const int buf = kc & 1;
        if (issuer) __builtin_amdgcn_s_wait_tensorcnt(0);
        __syncthreads();                       // chunk kc visible; buf^1 free
        if (issuer && (kc + 1 < nchunk)) issue_chunk(buf ^ 1, (kc + 1) * KC);

        const float* __restrict__ Ab = &Abuf[buf][0];
        const float* __restrict__ Bb = &Bbuf[buf][0];
#pragma unroll
        for (int kk = 0; kk < KC; kk += 4) {
            const int ka = kk + 2 * half;
            // A frag 16x4: lanes 0-15 hold K=kk,kk+1 ; lanes 16-31 K=kk+2,kk+3
            v2f a0 = *(const v2f*)&Ab[(wr * 32 + l)      * A_STRIDE + ka];
            v2f a1 = *(const v2f*)&Ab[(wr * 32 + 16 + l) * A_STRIDE + ka];
            v2f b0, b1;
            b0.x = Bb[(ka)     * B_STRIDE + wc * 32 + l];
            b0.y = Bb[(ka + 1) * B_STRIDE + wc * 32 + l];
            b1.x = Bb[(ka)     * B_STRIDE + wc * 32 + 16 + l];
            b1.y = Bb[(ka + 1) * B_STRIDE + wc * 32 + 16 + l];

            c00 = __builtin_amdgcn_wmma_f32_16x16x4_f32(false, a0, false, b0, (short)0, c00, false, false);
            c01 = __builtin_amdgcn_wmma_f32_16x16x4_f32(false, a0, false, b1, (short)0, c01, false, false);
            c10 = __builtin_amdgcn_wmma_f32_16x16x4_f32(false, a1, false, b0, (short)0, c10, false, false);
            c11 = __builtin_amdgcn_wmma_f32_16x16x4_f32(false, a1, false, b1, (short)0, c11, false, false);
        }
        __syncthreads();                       // done reading buf before refill
    }

    // C/D layout: VGPR r -> row r + 8*half, col = lane%16
    const int m0   = mblk + wr * 32;
    const int n0   = nblk + wc * 32;
    const int col0 = n0 + l;
    const int col1 = n0 + 16 + l;
#pragma unroll
    for (int r = 0; r < 8; ++r) {
        const int row0 = m0 + r + 8 * half;
        const int row1 = row0 + 16;
        if (col0 < N) {
            C[(size_t)row0 * ldc + col0] = c00[r];
            C[(size_t)row1 * ldc + col0] = c10[r];
        }
        if (col1 < N) {
            C[(size_t)row0 * ldc + col1] = c01[r];
            C[(size_t)row1 * ldc + col1] = c11[r];
        }
    }
}

// =====================================================================
// Depthwise causal conv(4) + SiLU on u; softplus(delta_pre+b)*exp(-a*s2)
// u_pre lives in xz[:, 0:2048]; delta computed in-place in delta_buf.
// =====================================================================
__global__ __launch_bounds__(256)
void conv_delta_kernel(const float* __restrict__ xz,
                       float* __restrict__ u_buf,
                       float* __restrict__ delta_buf,
                       const float* __restrict__ sigma2,
                       const float* __restrict__ conv_w,
                       const float* __restrict__ conv_b,
                       const float* __restrict__ b_delta,
                       const float* __restrict__ alpha_p)
{
    const int idx = blockIdx.x * 256 + threadIdx.x;     // < B*T*D_INNER
    const int i  = idx & (D_INNER - 1);
    const int bt = idx >> 11;                           // b*T + t
    const int t  = bt & (T_LEN - 1);

    float acc = conv_b[i];
#pragma unroll
    for (int j = 0; j < D_CONV; ++j) {
        const int ts = t + j - (D_CONV - 1);
        const float v = (ts >= 0) ? xz[(size_t)(bt + j - (D_CONV - 1)) * (2 * D_INNER) + i] : 0.0f;
        acc += conv_w[j * D_INNER + i] * v;
    }
    const float u = acc / (1.0f + __expf(-acc));        // SiLU
    u_buf[idx] = u;

    const float dp = delta_buf[idx] + b_delta[i];
    const float sp = (dp > 20.0f) ? dp : __logf(1.0f + __expf(dp));
    delta_buf[idx] = sp * __expf(-alpha_p[0] * sigma2[bt]);
}

// =====================================================================
// Selective scan: one thread per (batch, channel). 16 states + A row in
// registers; Bp/Cp loads are block-uniform (broadcast).
// Writes gated y in-place over u_buf.
// =====================================================================
__global__ __launch_bounds__(256)
void scan_kernel(const float* __restrict__ xz,        // z = xz[:, 2048:4096]
                 float* __restrict__ u_buf,           // in: u, out: y*silu(z)
                 const float* __restrict__ delta_buf,
                 const float* __restrict__ BpCp,      // (B*T, 32): [Bp|Cp]
                 const float* __restrict__ A_log,
                 const float* __restrict__ D_param)
{
    const int id = blockIdx.x * 256 + threadIdx.x;    // < B*D_INNER
    const int b  = id >> 11;
    const int i  = id & (D_INNER - 1);

    float Arow[D_STATE];
#pragma unroll
    for (int s = 0; s < D_STATE; ++s)
        Arow[s] = -__expf(A_log[i * D_STATE + s]);
    const float Dp = D_param[i];

    float h[D_STATE];
#pragma unroll
    for (int s = 0; s < D_STATE; ++s) h[s] = 0.0f;

    const size_t baseBT = (size_t)b * T_LEN;
    for (int t = 0; t < T_LEN; ++t) {
        const size_t bt = baseBT + t;
        const float d  = delta_buf[bt * D_INNER + i];
        const float uu = u_buf[bt * D_INNER + i];
        const float zz = xz[bt * (2 * D_INNER) + D_INNER + i];
        const float* __restrict__ bc = BpCp + bt * (2 * D_STATE);
        const float du = d * uu;

        float y = 0.0f;
#pragma unroll
        for (int s = 0; s < D_STATE; ++s) {
            const float dA = __expf(d * Arow[s]);
            h[s] = dA * h[s] + du * bc[s];
            y += h[s] * bc[D_STATE + s];
        }
        y += uu * Dp;
        const float sg = zz / (1.0f + __expf(-zz));   // silu(z)
        u_buf[bt * D_INNER + i] = y * sg;
    }
}

// =====================================================================
extern "C" void kernel_launch(void* const* d_in, const int* in_sizes, int n_in,
                              void* d_out, int out_size, void* d_ws, size_t ws_size,
                              hipStream_t stream)
{
    const float* x       = (const float*)d_in[0];
    const float* sigma2  = (const float*)d_in[1];
    const float* W_in    = (const float*)d_in[2];
    const float* conv_w  = (const float*)d_in[3];
    const float* conv_b  = (const float*)d_in[4];
    const float* W_xproj = (const float*)d_in[5];
    const float* W_delta = (const float*)d_in[6];
    const float* b_delta = (const float*)d_in[7];
    const float* A_log   = (const float*)d_in[8];
    const float* D_param = (const float*)d_in[9];
    const float* W_out   = (const float*)d_in[10];
    const float* alpha_p = (const float*)d_in[11];
    float* out = (float*)d_out;

    char* ws = (char*)d_ws;
    const size_t XZ_BYTES = (size_t)M_ROWS * 2 * D_INNER * 4;   // 64 MB
    const size_t DI_BYTES = (size_t)M_ROWS * D_INNER * 4;       // 32 MB
    float* xz    = (float*)(ws);
    float* delta = (float*)(ws + XZ_BYTES);
    float* u     = (float*)(ws + XZ_BYTES + DI_BYTES);
    float* BpCp  = (float*)(ws + XZ_BYTES + 2 * DI_BYTES);

    dim3 blk(256);

    // 1) xz = x @ W_in                (4096x1024 @ 1024x4096)
    wmma_gemm_f32<<<dim3(2 * D_INNER / BN, M_ROWS / BM), blk, 0, stream>>>(
        x, W_in, xz, M_ROWS, 2 * D_INNER, D_MODEL, D_MODEL, 2 * D_INNER, 2 * D_INNER);

    // 2) delta_pre = x @ W_delta      (4096x1024 @ 1024x2048)
    wmma_gemm_f32<<<dim3(D_INNER / BN, M_ROWS / BM), blk, 0, stream>>>(
        x, W_delta, delta, M_ROWS, D_INNER, D_MODEL, D_MODEL, D_INNER, D_INNER);

    // 3) conv + silu + delta gating
    conv_delta_kernel<<<dim3((M_ROWS * D_INNER) / 256), blk, 0, stream>>>(
        xz, u, delta, sigma2, conv_w, conv_b, b_delta, alpha_p);

    // 4) [Bp|Cp] = u @ W_xproj[:, :32]  (4096x2048 @ 2048x32, ldb=96)
    wmma_gemm_f32<<<dim3(1, M_ROWS / BM), blk, 0, stream>>>(
        u, W_xproj, BpCp, M_ROWS, 2 * D_STATE, D_INNER, D_INNER, 2 * D_STATE + 64, 2 * D_STATE);

    // 5) selective scan + skip + gate  (in-place into u)
    scan_kernel<<<dim3((B_SZ * D_INNER) / 256), blk, 0, stream>>>(
        xz, u, delta, BpCp, A_log, D_param);

    // 6) out = y @ W_out               (4096x2048 @ 2048x1024)
    wmma_gemm_f32<<<dim3(D_MODEL / BN, M_ROWS / BM), blk, 0, stream>>>(
        u, W_out, out, M_ROWS, D_MODEL, D_INNER, D_INNER, D_MODEL, D_MODEL);

    (void)in_sizes; (void)n_in; (void)out_size; (void)ws_size;
}